// DotProductAttention_62783831933126
// MI455X (gfx1250) — compile-verified
//
#include <hip/hip_runtime.h>
#include <hip/hip_bf16.h>

// CDNA5 fused attention: QK^T (f32 WMMA) -> softmax in LDS -> NT-stream weights
// -> PV (f32 WMMA) with deferred 1/sum scaling.

typedef __attribute__((ext_vector_type(2))) float v2f;
typedef __attribute__((ext_vector_type(4))) float v4f;
typedef __attribute__((ext_vector_type(8))) float v8f;

#define SEQ 2048
#define DH 64
#define ROWS 16            // query rows per workgroup
#define LDS_STRIDE 2052    // padded row stride in floats (2052 % 64 == 4 -> bank spread)

__global__ __launch_bounds__(256)
void attn_fused_f32(const float* __restrict__ Q,
                    const float* __restrict__ K,
                    const float* __restrict__ V,
                    float* __restrict__ Octx,
                    float* __restrict__ W) {
    __shared__ float sc[ROWS * LDS_STRIDE];   // 16 x 2048 exp-score strip (padded) ~128KB
    __shared__ float pvs[4 * 256];            // PV partial-accumulator scratch (4KB)
    __shared__ float rowinv[ROWS];            // per-row 1/sum

    const int lane = threadIdx.x & 31;
    const int wave = threadIdx.x >> 5;        // 0..7
    const int h    = lane >> 4;               // half-wave select (K-pair select)
    const int ln   = lane & 15;               // row/col within 16

    const int bid  = blockIdx.x;
    const int qblk = bid & 127;               // S/16 = 128 query blocks
    const int bh   = bid >> 7;                // 0..31 (b*H + h)

    const size_t head_base = (size_t)bh * SEQ * DH;
    const float* Qb = Q + head_base + (size_t)qblk * ROWS * DH;
    const float* Kb = K + head_base;
    const float* Vb = V + head_base;
    float* Wb = W + (size_t)bh * SEQ * SEQ + (size_t)qblk * ROWS * SEQ;
    float* Ob = Octx + head_base + (size_t)qblk * ROWS * DH;

    // ---------- preload Q block as A-fragments for all 16 d-chunks ----------
    // A (16x4 fp32): lane m=ln; VGPR0 = K{0|2}, VGPR1 = K{1|3} by half-wave.
    v2f aq[16];
    #pragma unroll
    for (int c = 0; c < 16; ++c) {
        aq[c] = *(const v2f*)(Qb + ln * DH + 4 * c + 2 * h);
    }

    // ---------- phase 1: scores = Q K^T into LDS ----------
    // wave w covers column tiles [w*16, w*16+16)
    for (int t = 0; t < 16; ++t) {
        const int n0 = (wave * 16 + t) * 16;
        const float* Krow = Kb + (size_t)(n0 + ln) * DH;   // B[k][n] = K[n][k]
        v8f acc = {};
        #pragma unroll
        for (int c = 0; c < 16; ++c) {
            v2f b = *(const v2f*)(Krow + 4 * c + 2 * h);
            acc = __builtin_amdgcn_wmma_f32_16x16x4_f32(
                false, aq[c], false, b, (short)0, acc, false, false);
        }
        // C layout: VGPR r -> (M = r + 8*h, N = n0 + ln)
        #pragma unroll
        for (int r = 0; r < 8; ++r) {
            sc[(r + 8 * h) * LDS_STRIDE + n0 + ln] = acc[r];
        }
    }
    __syncthreads();

    // ---------- phase 2: per-row softmax; NT-stream weights; keep e in LDS ----------
    for (int rr = 0; rr < 2; ++rr) {
        const int row = wave * 2 + rr;
        float* srow = sc + row * LDS_STRIDE;

        // row max (vectorized b128 LDS reads)
        float mx = -3.402823466e+38f;
        for (int k2 = lane * 4; k2 < SEQ; k2 += 128) {
            v4f x = *(const v4f*)(srow + k2);
            mx = fmaxf(mx, fmaxf(fmaxf(x.x, x.y), fmaxf(x.z, x.w)));
        }
        #pragma unroll
        for (int off = 16; off > 0; off >>= 1) mx = fmaxf(mx, __shfl_xor(mx, off, 32));

        // exp + running sum; store e (un-normalized) back to LDS
        float s = 0.0f;
        for (int k2 = lane * 4; k2 < SEQ; k2 += 128) {
            v4f x = *(const v4f*)(srow + k2);
            v4f e;
            e.x = __expf(x.x - mx);
            e.y = __expf(x.y - mx);
            e.z = __expf(x.z - mx);
            e.w = __expf(x.w - mx);
            *(v4f*)(srow + k2) = e;
            s += (e.x + e.y) + (e.z + e.w);
        }
        #pragma unroll
        for (int off = 16; off > 0; off >>= 1) s += __shfl_xor(s, off, 32);
        const float inv = 1.0f / s;
        if (lane == 0) rowinv[row] = inv;

        // stream normalized weights to HBM, non-temporal (write-once stream;
        // keep K/V resident in L2 instead)
        float* wrow = Wb + (size_t)row * SEQ;
        for (int k2 = lane * 4; k2 < SEQ; k2 += 128) {
            v4f e = *(const v4f*)(srow + k2);
            v4f p = { e.x * inv, e.y * inv, e.z * inv, e.w * inv };
            __builtin_nontemporal_store(p, (v4f*)(wrow + k2));
        }
    }
    __syncthreads();

    // ---------- phase 3: O = inv[m] * (E V) ----------
    // wave (w&3) owns output d-tile, (w>>2) owns a 1024-wide K half.
    const int dt    = wave & 3;
    const int d0    = dt * 16;
    const int kbase = (wave >> 2) * 1024;

    v8f acc = {};
    for (int k0 = kbase; k0 < kbase + 1024; k0 += 4) {
        // A (16x4): E[m][k0+2h], E[m][k0+2h+1] from LDS (bank-padded column walk)
        v2f a = *(const v2f*)(sc + ln * LDS_STRIDE + k0 + 2 * h);
        // B (4x16): V[k][d0+n]
        v2f b;
        b.x = Vb[(size_t)(k0 + 2 * h)     * DH + d0 + ln];
        b.y = Vb[(size_t)(k0 + 2 * h + 1) * DH + d0 + ln];
        acc = __builtin_amdgcn_wmma_f32_16x16x4_f32(
            false, a, false, b, (short)0, acc, false, false);
    }

    if (wave >= 4) {
        #pragma unroll
        for (int r = 0; r < 8; ++r)
            pvs[dt * 256 + (r + 8 * h) * 16 + ln] = acc[r];
    }
    __syncthreads();
    if (wave < 4) {
        #pragma unroll
        for (int r = 0; r < 8; ++r) {
            const int m = r + 8 * h;
            float o = (acc[r] + pvs[dt * 256 + m * 16 + ln]) * rowinv[m];
            Ob[(size_t)m * DH + d0 + ln] = o;
        }
    }
}

extern "C" void kernel_launch(void* const* d_in, const int* in_sizes, int n_in,
                              void* d_out, int out_size, void* d_ws, size_t ws_size,
                              hipStream_t stream) {
    (void)in_sizes; (void)n_in; (void)out_size; (void)d_ws; (void)ws_size;
    const float* Q = (const float*)d_in[0];
    const float* K = (const float*)d_in[1];
    const float* V = (const float*)d_in[2];

    const size_t B = 2, H = 16, S = 2048, D = 64;
    float* ctx = (float*)d_out;                       // [B,H,S,D]
    float* wts = (float*)d_out + B * H * S * D;       // [B,H,S,S]

    dim3 grid((unsigned)(B * H * (S / ROWS)));        // 4096 workgroups
    dim3 block(256);                                  // 8 wave32
    hipLaunchKernelGGL(attn_fused_f32, grid, block, 0, stream, Q, K, V, ctx, wts);
}